// SelectivePool_14534169330327
// MI455X (gfx1250) — compile-verified
//
#include <hip/hip_runtime.h>

typedef float v2f __attribute__((ext_vector_type(2)));
typedef float v8f __attribute__((ext_vector_type(8)));

#define BB   32
#define CC   768
#define HIDC 96
#define HH   32
#define WWD  32
#define NN   1024          // HH*WWD
#define KSEL 256
#define BN_EPS 1e-5f

// ---------------------------------------------------------------------------
// Kernel A: fused 1x1-conv GEMM (V_WMMA_F32_16X16X4_F32) + BatchNorm + ReLU
//   x  : [B, C, N]  (N = H*W)
//   w1 : [HID, C]
//   h  : [B*N, HID]   (position-major so the 3x3 conv reads contiguous vectors)
// One wave handles a 16-position row tile across all 96 HID columns.
// ---------------------------------------------------------------------------
__global__ __launch_bounds__(256) void gemm_bn_relu_kernel(
    const float* __restrict__ x, const float* __restrict__ w1,
    const float* __restrict__ gamma, const float* __restrict__ beta,
    const float* __restrict__ mean,  const float* __restrict__ var,
    float* __restrict__ h)
{
    const int lane = threadIdx.x & 31;
    const int wave = threadIdx.x >> 5;
    const int tile = blockIdx.x * 8 + wave;     // 2048 tiles of 16 positions
    const int p0   = tile * 16;
    const int b    = p0 >> 10;                  // N == 1024, tiles never cross batch
    const int n0   = p0 & (NN - 1);
    const int half = lane >> 4;
    const int l    = lane & 15;

    const float* xb = x + (size_t)b * CC * NN;

    v8f acc[6] = {};

    for (int k = 0; k < CC; k += 4) {
        // A fragment 16x4 f32: lanes 0-15 -> K = k,k+1 ; lanes 16-31 -> K = k+2,k+3
        v2f a;
        a.x = xb[(size_t)(k + 2 * half + 0) * NN + (n0 + l)];
        a.y = xb[(size_t)(k + 2 * half + 1) * NN + (n0 + l)];
#pragma unroll
        for (int t = 0; t < 6; ++t) {
            const int o = t * 16 + l;           // B fragment 4x16: lane = N(=o)
            v2f bf;
            bf.x = w1[(size_t)o * CC + (k + 2 * half + 0)];
            bf.y = w1[(size_t)o * CC + (k + 2 * half + 1)];
            acc[t] = __builtin_amdgcn_wmma_f32_16x16x4_f32(
                false, a, false, bf, (short)0, acc[t], false, false);
        }
    }

    // BN (inference) + ReLU epilogue, store h[pos][o]
#pragma unroll
    for (int t = 0; t < 6; ++t) {
        const int o = t * 16 + l;
        const float inv = rsqrtf(var[o] + BN_EPS);
        const float sc  = gamma[o] * inv;
        const float bi  = beta[o] - mean[o] * sc;
#pragma unroll
        for (int i = 0; i < 8; ++i) {
            const int m = i + 8 * half;         // C/D layout: VGPR i -> M=i / M=i+8
            float vv = acc[t][i] * sc + bi;
            vv = vv > 0.f ? vv : 0.f;
            h[(size_t)(p0 + m) * HIDC + o] = vv;
        }
    }
}

// ---------------------------------------------------------------------------
// Kernel B: 3x3 conv HID->1, SAME padding. Thread per output pixel.
//   h     : [B*N, HID]
//   score : [B*N]  (written into d_out tail)
// ---------------------------------------------------------------------------
__global__ __launch_bounds__(256) void conv3_kernel(
    const float* __restrict__ h, const float* __restrict__ w2,
    const float* __restrict__ b2, float* __restrict__ score)
{
    __shared__ float w2s[9][HIDC];              // [tap][o], rows 384B -> 16B aligned
    for (int i = threadIdx.x; i < 9 * HIDC; i += 256) {
        const int o = i / 9, j = i % 9;         // w2 original layout: [o*9 + dy*3 + dx]
        w2s[j][o] = w2[i];
    }
    __syncthreads();

    const int pix = blockIdx.x * 256 + threadIdx.x;   // 32768 pixels
    const int b   = pix >> 10;
    const int yx  = pix & (NN - 1);
    const int y   = yx >> 5;
    const int xw  = yx & 31;

    float acc = b2[0];
    for (int dy = -1; dy <= 1; ++dy) {
        const int yy = y + dy;
        if (yy < 0 || yy >= HH) continue;
        for (int dx = -1; dx <= 1; ++dx) {
            const int xx = xw + dx;
            if (xx < 0 || xx >= WWD) continue;
            const float* hp = h + (size_t)(b * NN + yy * WWD + xx) * HIDC;
            const float* wp = &w2s[(dy + 1) * 3 + (dx + 1)][0];
            float s = 0.f;
#pragma unroll
            for (int o = 0; o < HIDC; o += 4) {
                const float4 hv = *(const float4*)(hp + o);
                const float4 wv = *(const float4*)(wp + o);
                s = fmaf(hv.x, wv.x, s);
                s = fmaf(hv.y, wv.y, s);
                s = fmaf(hv.z, wv.z, s);
                s = fmaf(hv.w, wv.w, s);
            }
            acc += s;
        }
    }
    score[pix] = acc;
}

// ---------------------------------------------------------------------------
// Kernel C: per-batch exact top-256 (rank counting, stable tie-break matching
// jax.lax.top_k), softmax over selected, weighted gather-reduce of x.
// One workgroup (256 threads) per batch. x re-reads hit L2 (192MB >= 100MB).
// ---------------------------------------------------------------------------
__global__ __launch_bounds__(256) void topk_pool_kernel(
    const float* __restrict__ x, const float* __restrict__ score,
    float* __restrict__ v)
{
    __shared__ float s[NN];
    __shared__ float red[256];
    __shared__ int   idxs[KSEL];
    __shared__ float ws[KSEL];
    __shared__ int   cnt;

    const int b = blockIdx.x;
    const int t = threadIdx.x;
    const float* sb = score + b * NN;

    for (int i = t; i < NN; i += 256) s[i] = sb[i];
    if (t == 0) cnt = 0;
    __syncthreads();

    // global max (top-1 is always selected, so softmax max == global max)
    float mx = -1e30f;
    for (int i = t; i < NN; i += 256) mx = fmaxf(mx, s[i]);
    red[t] = mx;
    __syncthreads();
    for (int off = 128; off > 0; off >>= 1) {
        if (t < off) red[t] = fmaxf(red[t], red[t + off]);
        __syncthreads();
    }
    mx = red[0];
    __syncthreads();

    // rank each of my 4 candidates; selected iff rank < KSEL
    int   myIdx[4]; float myE[4]; int mySel[4];
    float esum = 0.f;
#pragma unroll
    for (int q = 0; q < 4; ++q) {
        const int   n  = t + q * 256;
        const float sn = s[n];
        int rank = 0;
        for (int j = 0; j < NN; ++j) {
            const float sj = s[j];
            rank += (sj > sn) || ((sj == sn) && (j < n));
        }
        const int   sel = rank < KSEL;
        const float e   = sel ? __expf(sn - mx) : 0.f;
        myIdx[q] = n; myE[q] = e; mySel[q] = sel;
        esum += e;
    }
    red[t] = esum;
    __syncthreads();
    for (int off = 128; off > 0; off >>= 1) {
        if (t < off) red[t] += red[t + off];
        __syncthreads();
    }
    const float inv = 1.f / red[0];
    __syncthreads();

    // compact selected (idx, weight) pairs; order irrelevant for the sum
#pragma unroll
    for (int q = 0; q < 4; ++q) {
        if (mySel[q]) {
            const int pos = atomicAdd(&cnt, 1);
            idxs[pos] = myIdx[q];
            ws[pos]   = myE[q] * inv;
        }
    }
    __syncthreads();

    // v[b,c] = sum_k ws[k] * x[b,c,idxs[k]]   (3 channels per thread)
    const float* xb = x + (size_t)b * CC * NN;
    float a0 = 0.f, a1 = 0.f, a2 = 0.f;
    for (int k = 0; k < KSEL; ++k) {
        const int   n  = idxs[k];
        const float wk = ws[k];
        a0 = fmaf(wk, xb[(size_t)(t      ) * NN + n], a0);
        a1 = fmaf(wk, xb[(size_t)(t + 256) * NN + n], a1);
        a2 = fmaf(wk, xb[(size_t)(t + 512) * NN + n], a2);
    }
    v[b * CC + t      ] = a0;
    v[b * CC + t + 256] = a1;
    v[b * CC + t + 512] = a2;
}

// ---------------------------------------------------------------------------
extern "C" void kernel_launch(void* const* d_in, const int* in_sizes, int n_in,
                              void* d_out, int out_size, void* d_ws, size_t ws_size,
                              hipStream_t stream) {
    const float* x     = (const float*)d_in[0];   // [32,768,32,32]
    const float* w1    = (const float*)d_in[1];   // [96,768,1,1]
    const float* gamma = (const float*)d_in[2];   // [96]
    const float* beta  = (const float*)d_in[3];   // [96]
    const float* mean  = (const float*)d_in[4];   // [96]
    const float* var   = (const float*)d_in[5];   // [96]
    const float* w2    = (const float*)d_in[6];   // [1,96,3,3]
    const float* b2    = (const float*)d_in[7];   // [1]

    float* out   = (float*)d_out;
    float* v     = out;                 // [32,768]      -> 24576 floats
    float* score = out + BB * CC;       // [32,1,32,32]  -> 32768 floats
    float* h     = (float*)d_ws;        // [32768, 96]   -> 12.6 MB scratch

    // A: GEMM(+BN+ReLU): 2048 row-tiles, 8 waves/block -> 256 blocks
    gemm_bn_relu_kernel<<<256, 256, 0, stream>>>(x, w1, gamma, beta, mean, var, h);
    // B: 3x3 conv -> score
    conv3_kernel<<<128, 256, 0, stream>>>(h, w2, b2, score);
    // C: top-k + softmax + weighted pool -> v
    topk_pool_kernel<<<32, 256, 0, stream>>>(x, score, v);
}